// HabanaEmbeddingBag_74904229642565
// MI455X (gfx1250) — compile-verified
//
#include <hip/hip_runtime.h>

// EmbeddingBag(sum): out[b][d] = sum_{i in bag b} W[indices[i]][d]
// W: [1e6 x 64] f32, indices: [819200] i32, offsets: [16384] i32 -> out [16384 x 64] f32
//
// Strategy: one wave32 per bag. Wave-uniform index fetches (s_load via readfirstlane'd
// bag id). Row accumulation done on the matrix pipe: V_WMMA_F32_16X16X4_F32 with
// A = ones(16x4) computes D[m][n] = sum_k B[k][n] + C[m][n], i.e. adds 4 gathered
// rows into 16 column sums per instruction. 4 dim-tiles cover EMB_DIM=64.

typedef __attribute__((ext_vector_type(2))) float v2f;
typedef __attribute__((ext_vector_type(8))) float v8f;

#define WAVES_PER_BLOCK 8
#define EMB_DIM 64

__global__ __launch_bounds__(256) void embbag_wmma_kernel(
    const float* __restrict__ W,
    const int*   __restrict__ indices,
    const int*   __restrict__ offsets,
    float*       __restrict__ out,
    int n_bags, int n_idx)
{
    // wave id within block -> force scalar so index loads become s_load (KMcnt path)
    const int wave = __builtin_amdgcn_readfirstlane((int)(threadIdx.x >> 5));
    const int lane = (int)(threadIdx.x & 31);
    const int bag  = (int)blockIdx.x * WAVES_PER_BLOCK + wave;
    if (bag >= n_bags) return;   // wave-uniform branch: EXEC stays all-ones for WMMA

    const int start = offsets[bag];
    const int end   = (bag + 1 < n_bags) ? offsets[bag + 1] : n_idx;
    const int count = end - start;

    const int  nsub   = lane & 15;     // N (dim within 16-wide tile) = lane % 16
    const bool hiHalf = lane >= 16;    // upper lane half carries the other 2 K-rows

    // 4 accumulator tiles: dims [0,16) [16,32) [32,48) [48,64)
    v8f c0 = {}, c1 = {}, c2 = {}, c3 = {};
    v2f ones; ones[0] = 1.0f; ones[1] = 1.0f;   // A = all-ones 16x4

    const int count4 = count & ~3;
    int j = 0;
#pragma unroll 2
    for (; j < count4; j += 4) {
        // wave-uniform row indices -> scalar loads
        const int r0 = indices[start + j + 0];
        const int r1 = indices[start + j + 1];
        const int r2 = indices[start + j + 2];
        const int r3 = indices[start + j + 3];
        // lanes 0-15 fetch rows r0/r1, lanes 16-31 fetch rows r2/r3;
        // each global_load_b32 covers two 64B-aligned contiguous segments.
        const float* rowA = W + (size_t)(hiHalf ? r2 : r0) * EMB_DIM;
        const float* rowB = W + (size_t)(hiHalf ? r3 : r1) * EMB_DIM;
        v2f b0, b1, b2, b3;
        b0[0] = rowA[ 0 + nsub];  b0[1] = rowB[ 0 + nsub];
        b1[0] = rowA[16 + nsub];  b1[1] = rowB[16 + nsub];
        b2[0] = rowA[32 + nsub];  b2[1] = rowB[32 + nsub];
        b3[0] = rowA[48 + nsub];  b3[1] = rowB[48 + nsub];
        // D = ones(16x4) x B(4x16) + C : accumulates 4 rows into 16 column sums.
        // A all-ones makes the result invariant to the K<->register permutation in B.
        c0 = __builtin_amdgcn_wmma_f32_16x16x4_f32(false, ones, false, b0, (short)0, c0, false, false);
        c1 = __builtin_amdgcn_wmma_f32_16x16x4_f32(false, ones, false, b1, (short)0, c1, false, false);
        c2 = __builtin_amdgcn_wmma_f32_16x16x4_f32(false, ones, false, b2, (short)0, c2, false, false);
        c3 = __builtin_amdgcn_wmma_f32_16x16x4_f32(false, ones, false, b3, (short)0, c3, false, false);
    }
    // remainder rows (count % 4): plain VALU adds into the D-row slot we read out
    // (c[0] == row m=0 for lanes 0-15, row m=8 for lanes 16-31; both rows are equal).
    for (; j < count; ++j) {
        const int r = indices[start + j];
        const float* row = W + (size_t)r * EMB_DIM;
        c0[0] += row[ 0 + nsub];
        c1[0] += row[16 + nsub];
        c2[0] += row[32 + nsub];
        c3[0] += row[48 + nsub];
    }

    // c*[0] holds colsum for dim (tile*16 + nsub) in every lane; split the 4 tiles
    // across the two lane halves -> 2 coalesced b32 stores per wave.
    float* o = out + (size_t)bag * EMB_DIM;
    if (!hiHalf) {
        o[ 0 + nsub] = c0[0];
        o[16 + nsub] = c1[0];
    } else {
        o[32 + nsub] = c2[0];
        o[48 + nsub] = c3[0];
    }
}

extern "C" void kernel_launch(void* const* d_in, const int* in_sizes, int n_in,
                              void* d_out, int out_size, void* d_ws, size_t ws_size,
                              hipStream_t stream) {
    const float* W       = (const float*)d_in[0];
    const int*   indices = (const int*)d_in[1];
    const int*   offsets = (const int*)d_in[2];
    float*       out     = (float*)d_out;

    const int n_idx  = in_sizes[1];
    const int n_bags = in_sizes[2];

    const int blocks = (n_bags + WAVES_PER_BLOCK - 1) / WAVES_PER_BLOCK;
    embbag_wmma_kernel<<<blocks, WAVES_PER_BLOCK * 32, 0, stream>>>(
        W, indices, offsets, out, n_bags, n_idx);
}